// OMul_17746804867414
// MI455X (gfx1250) — compile-verified
//
#include <hip/hip_runtime.h>
#include <hip/hip_bf16.h>

// Batched outer product r[i] = x[i] (x) y[i] for x,y:[N,256] f32 -> out:[N,65536] f32.
// Store-bandwidth bound (1 GiB out @ 23.3 TB/s ~ 46 us). Tiles generated with
// V_WMMA_F32_16X16X4_F32 as a rank-1 (K=1) matmul; NT stores stream past L2.
//
// WMMA K-slot layout (f32 16x16x4): K=0 -> VGPR0 lanes 0-15, K=1 -> VGPR1 lanes 0-15,
// K=2 -> VGPR0 lanes 16-31, K=3 -> VGPR1 lanes 16-31.  Zeroing K=1..3 on the B
// operand alone makes A's non-K0 slots don't-care, so A is loaded unmasked.

typedef __attribute__((ext_vector_type(2))) float v2f;
typedef __attribute__((ext_vector_type(8))) float v8f;

#define OM_M 256
#define OM_K 256

__global__ __launch_bounds__(256) void OMul_wmma_kernel(
    const float* __restrict__ x,
    const float* __restrict__ y,
    float* __restrict__ out)
{
    __shared__ float sx[OM_M];
    __shared__ float sy[OM_K];

    const int i = blockIdx.x;      // batch element
    const int t = threadIdx.x;     // 0..255

    // Stage x[i] and y[i] (2 KB total) into LDS once.
    sx[t] = x[(size_t)i * OM_M + t];
    sy[t] = y[(size_t)i * OM_K + t];
    __syncthreads();

    const int wave = t >> 5;       // 0..7 (wave32)
    const int lane = t & 31;
    const int l16  = lane & 15;
    const bool lo  = (lane < 16);

    float* const base_out = out + (size_t)i * (OM_M * OM_K);

    // Each wave: 2 tile-rows of 16 tiles = 32 WMMA 16x16 tiles.
    #pragma unroll
    for (int tmi = 0; tmi < 2; ++tmi) {
        const int tm = wave * 2 + tmi;

        // A (16x4 f32): unconditional LDS load; lanes 16-31 land in the K=2
        // slot which is multiplied by B's zeroed K=2 row -> don't-care.
        v2f a;
        a.x = sx[tm * 16 + l16];
        a.y = 0.0f;

        #pragma unroll
        for (int tn = 0; tn < 16; ++tn) {
            // B (4x16 f32): row K=0 = y values in lanes 0-15; all other K
            // rows zero.  Mask applied to the value (v_cndmask), not the load.
            const float yv = sy[tn * 16 + l16];
            v2f b;
            b.x = lo ? yv : 0.0f;
            b.y = 0.0f;

            v8f c = {};
            // 8 args: (neg_a, A, neg_b, B, c_mod, C, reuse_a, reuse_b)
            v8f d = __builtin_amdgcn_wmma_f32_16x16x4_f32(
                false, a, false, b, (short)0, c, false, false);

            // D layout: VGPR v -> lanes 0-15: (M=v,   N=lane),
            //                     lanes 16-31:(M=v+8, N=lane-16)
            float* const tile = base_out + (size_t)(tm * 16) * OM_K + tn * 16;
            const int row_off = lo ? 0 : 8;
            #pragma unroll
            for (int v = 0; v < 8; ++v) {
                __builtin_nontemporal_store(
                    d[v], tile + (size_t)(v + row_off) * OM_K + l16);
            }
        }
    }
}

extern "C" void kernel_launch(void* const* d_in, const int* in_sizes, int n_in,
                              void* d_out, int out_size, void* d_ws, size_t ws_size,
                              hipStream_t stream) {
    const float* x = (const float*)d_in[0];
    const float* y = (const float*)d_in[1];
    float* out = (float*)d_out;

    const int n = in_sizes[0] / OM_M;   // 4096 batch elements
    OMul_wmma_kernel<<<n, 256, 0, stream>>>(x, y, out);
}